// RotationLaplaceLoss_3332894622420
// MI455X (gfx1250) — compile-verified
//
#include <hip/hip_runtime.h>
#include <math.h>

typedef __attribute__((ext_vector_type(2))) float v2f;
typedef __attribute__((ext_vector_type(8))) float v8f;

#define BATCH  256
#define NGRID  36864
#define KPAD   12
#define CHUNKS 144            // NGRID / 256 columns per wave-task
#define TILES_PER_CHUNK 16    // 16 cols per WMMA tile * 16 tiles = 256 cols
#define EPS_TRACE 1e-8f

// ---------------- helpers ----------------

__device__ __forceinline__ double shflxor_f64(double v, int mask) {
    long long x = __double_as_longlong(v);
    int lo = (int)(x & 0xffffffffLL);
    int hi = (int)(x >> 32);
    lo = __shfl_xor(lo, mask, 32);
    hi = __shfl_xor(hi, mask, 32);
    return __longlong_as_double(((long long)hi << 32) | (unsigned long long)(unsigned int)lo);
}

__device__ __forceinline__ void quat2mat(const float* q, float R[3][3]) {
    float w = q[0], x = q[1], y = q[2], z = q[3];
    float inv = rsqrtf(w * w + x * x + y * y + z * z);
    w *= inv; x *= inv; y *= inv; z *= inv;
    R[0][0] = 1.f - 2.f * (y * y + z * z);
    R[0][1] = 2.f * (x * y - w * z);
    R[0][2] = 2.f * (x * z + w * y);
    R[1][0] = 2.f * (x * y + w * z);
    R[1][1] = 1.f - 2.f * (x * x + z * z);
    R[1][2] = 2.f * (y * z - w * x);
    R[2][0] = 2.f * (x * z - w * y);
    R[2][1] = 2.f * (y * z + w * x);
    R[2][2] = 1.f - 2.f * (x * x + y * y);
}

// online stabilized merge of (m1,S1) with (m2,S2), where S = sum exp(m - p)/p
__device__ __forceinline__ void lse_merge(float& m, double& S, float om, double oS) {
    float nm = fminf(m, om);
    S = S * exp((double)(nm - m)) + oS * exp((double)(nm - om));
    m = nm;
}

// ---------------- kernel 1: per-batch setup ----------------
// Z = (L L^T)^{-1} = Linv^T Linv (3x3 lower-tri inverse in closed form),
// tr(Z), power at ground truth, and Z flattened row-major padded to K=12.

__global__ void ksetup(const float* __restrict__ gt_quat,
                       const float* __restrict__ mode_quat,
                       const float* __restrict__ cov_tril,
                       float* __restrict__ Aarr,     // [BATCH][KPAD]
                       float* __restrict__ trZ,      // [BATCH]
                       float* __restrict__ powerGT)  // [BATCH]
{
    int b = blockIdx.x * blockDim.x + threadIdx.x;
    if (b >= BATCH) return;

    const float* L = cov_tril + b * 9;
    float l00 = L[0], l10 = L[3], l11 = L[4], l20 = L[6], l21 = L[7], l22 = L[8];
    float i00 = 1.f / l00, i11 = 1.f / l11, i22 = 1.f / l22;
    float i10 = -l10 * i00 * i11;
    float i21 = -l21 * i11 * i22;
    float i20 = (l10 * l21 - l11 * l20) * i00 * i11 * i22;
    float Li[3][3] = {{i00, 0.f, 0.f}, {i10, i11, 0.f}, {i20, i21, i22}};

    float Z[3][3];
    #pragma unroll
    for (int i = 0; i < 3; ++i)
        #pragma unroll
        for (int j = 0; j < 3; ++j) {
            float s = 0.f;
            #pragma unroll
            for (int k = 0; k < 3; ++k) s += Li[k][i] * Li[k][j];
            Z[i][j] = s;
        }
    float tr = Z[0][0] + Z[1][1] + Z[2][2];

    float Rg[3][3], R0[3][3];
    quat2mat(gt_quat + b * 4, Rg);
    quat2mat(mode_quat + b * 4, R0);

    // tr(Z @ Rrel) with Rrel = R0^T Rg : sum_ij Z[i][j] * Rrel[j][i]
    float trZR = 0.f;
    #pragma unroll
    for (int i = 0; i < 3; ++i)
        #pragma unroll
        for (int j = 0; j < 3; ++j) {
            float rr = 0.f; // Rrel[j][i]
            #pragma unroll
            for (int k = 0; k < 3; ++k) rr += R0[k][j] * Rg[k][i];
            trZR += Z[i][j] * rr;
        }

    float pg = sqrtf(fmaxf(tr - trZR, EPS_TRACE));

    #pragma unroll
    for (int k = 0; k < KPAD; ++k)
        Aarr[b * KPAD + k] = (k < 9) ? Z[k / 3][k % 3] : 0.f;
    trZ[b] = tr;
    powerGT[b] = pg;
}

// ---------------- kernel 2: repack grids into B-operand [KPAD][NGRID] ----------------
// tr(Z R_n) = sum_{i,j} Z[i,j] * R_n[j,i] -> B[k=i*3+j][n] = grids[n, j, i]

__global__ void ktranspose(const float* __restrict__ grids, float* __restrict__ Bg) {
    int idx = blockIdx.x * blockDim.x + threadIdx.x;
    if (idx >= KPAD * NGRID) return;
    int k = idx / NGRID;
    int n = idx - k * NGRID;
    Bg[idx] = (k < 9) ? grids[n * 9 + (k % 3) * 3 + (k / 3)] : 0.f;
}

// ---------------- kernel 3: fused WMMA GEMM + online stabilized reduction ----------------
// One wave computes a 16-row x 256-col strip of powers[B,N] (16 tiles of 16x16
// via V_WMMA_F32_16X16X4_F32, K=12 in three k-steps) and reduces it to one
// (min, stabilized-sum) pair per row without ever touching memory.

__global__ void __launch_bounds__(256)
kpowers(const float* __restrict__ Aarr, const float* __restrict__ Bg,
        const float* __restrict__ trZ,
        float* __restrict__ partM,   // [BATCH][CHUNKS]
        double* __restrict__ partS)  // [BATCH][CHUNKS]
{
    const int wave    = (blockIdx.x * blockDim.x + threadIdx.x) >> 5;
    const int lane    = threadIdx.x & 31;
    const int rowTile = wave / CHUNKS;
    const int chunk   = wave - rowTile * CHUNKS;
    const int m0      = rowTile * 16;
    const int half    = lane >> 4;     // 0: lanes 0-15, 1: lanes 16-31
    const int ln      = lane & 15;

    // A fragments: 16x4 f32 layout -> lane (M=ln) holds K=kk+2*half, kk+2*half+1
    v2f a0, a1, a2;
    {
        const float* ap = Aarr + (m0 + ln) * KPAD;
        a0.x = ap[0 + 2 * half]; a0.y = ap[1 + 2 * half];
        a1.x = ap[4 + 2 * half]; a1.y = ap[5 + 2 * half];
        a2.x = ap[8 + 2 * half]; a2.y = ap[9 + 2 * half];
    }

    // tr(Z) for the 8 rows this lane's accumulator covers (M = v + 8*half)
    float tz[8];
    #pragma unroll
    for (int v = 0; v < 8; ++v) tz[v] = trZ[m0 + 8 * half + v];

    float  mrun[8];
    double srun[8];
    #pragma unroll
    for (int v = 0; v < 8; ++v) { mrun[v] = 3.0e38f; srun[v] = 0.0; }

    const int nbase = chunk * 256;
    for (int t = 0; t < TILES_PER_CHUNK; ++t) {
        const int col = nbase + t * 16 + ln;   // N = ln within tile
        v2f b0, b1, b2;                        // B 4x16: lane holds K=kk+2*half(+1), N=ln
        b0.x = Bg[(0 + 2 * half) * NGRID + col]; b0.y = Bg[(1 + 2 * half) * NGRID + col];
        b1.x = Bg[(4 + 2 * half) * NGRID + col]; b1.y = Bg[(5 + 2 * half) * NGRID + col];
        b2.x = Bg[(8 + 2 * half) * NGRID + col]; b2.y = Bg[(9 + 2 * half) * NGRID + col];

        v8f c = {};
        c = __builtin_amdgcn_wmma_f32_16x16x4_f32(false, a0, false, b0, (short)0, c, false, false);
        c = __builtin_amdgcn_wmma_f32_16x16x4_f32(false, a1, false, b1, (short)0, c, false, false);
        c = __builtin_amdgcn_wmma_f32_16x16x4_f32(false, a2, false, b2, (short)0, c, false, false);

        #pragma unroll
        for (int v = 0; v < 8; ++v) {
            float p = sqrtf(fmaxf(tz[v] - c[v], EPS_TRACE));
            if (p < mrun[v]) {
                srun[v] = srun[v] * exp((double)(p - mrun[v])) + 1.0 / (double)p;
                mrun[v] = p;
            } else {
                srun[v] += exp((double)(mrun[v] - p)) / (double)p;
            }
        }
    }

    // reduce across the 16 columns: butterfly stays inside each 16-lane half
    #pragma unroll
    for (int v = 0; v < 8; ++v) {
        float  m = mrun[v];
        double S = srun[v];
        #pragma unroll
        for (int mask = 1; mask < 16; mask <<= 1) {
            float  om = __shfl_xor(m, mask, 32);
            double oS = shflxor_f64(S, mask);
            lse_merge(m, S, om, oS);
        }
        mrun[v] = m; srun[v] = S;
    }

    if (ln == 0) {
        #pragma unroll
        for (int v = 0; v < 8; ++v) {
            int row = m0 + 8 * half + v;
            partM[row * CHUNKS + chunk] = mrun[v];
            partS[row * CHUNKS + chunk] = srun[v];
        }
    }
}

// ---------------- kernel 4: per-batch final reduction ----------------

__global__ void __launch_bounds__(256)
kfinal(const float* __restrict__ partM, const double* __restrict__ partS,
       const float* __restrict__ powerGT, float* __restrict__ out)
{
    const int b    = (blockIdx.x * blockDim.x + threadIdx.x) >> 5;
    const int lane = threadIdx.x & 31;
    if (b >= BATCH) return;

    float  m = 3.0e38f;
    double S = 0.0;
    for (int c = lane; c < CHUNKS; c += 32)
        lse_merge(m, S, partM[b * CHUNKS + c], partS[b * CHUNKS + c]);

    #pragma unroll
    for (int mask = 1; mask < 32; mask <<= 1) {
        float  om = __shfl_xor(m, mask, 32);
        double oS = shflxor_f64(S, mask);
        lse_merge(m, S, om, oS);
    }

    if (lane == 0) {
        float log_exp_sum = (float)log(S);                       // fp64 accumulate, fp32 cast (ref)
        float logF = log_exp_sum - logf((float)NGRID) - m;
        float p = powerGT[b];
        out[b] = logF + p + logf(p);
    }
}

// ---------------- launcher ----------------

extern "C" void kernel_launch(void* const* d_in, const int* in_sizes, int n_in,
                              void* d_out, int out_size, void* d_ws, size_t ws_size,
                              hipStream_t stream) {
    const float* gt_quat   = (const float*)d_in[0];
    const float* mode_quat = (const float*)d_in[1];
    const float* cov_tril  = (const float*)d_in[2];
    const float* grids     = (const float*)d_in[3];
    float* out = (float*)d_out;

    char* ws = (char*)d_ws;
    // 8-byte aligned doubles first
    double* partS  = (double*)ws;                                   // 256*144*8   = 294912
    size_t off = (size_t)BATCH * CHUNKS * sizeof(double);
    float* Bg      = (float*)(ws + off);  off += (size_t)KPAD * NGRID * sizeof(float);   // 1769472
    float* Aarr    = (float*)(ws + off);  off += (size_t)BATCH * KPAD * sizeof(float);
    float* trZ     = (float*)(ws + off);  off += (size_t)BATCH * sizeof(float);
    float* powerGT = (float*)(ws + off);  off += (size_t)BATCH * sizeof(float);
    float* partM   = (float*)(ws + off);  off += (size_t)BATCH * CHUNKS * sizeof(float);
    (void)ws_size; (void)in_sizes; (void)n_in; (void)out_size;

    ksetup<<<1, 256, 0, stream>>>(gt_quat, mode_quat, cov_tril, Aarr, trZ, powerGT);

    int nrep = KPAD * NGRID;
    ktranspose<<<(nrep + 255) / 256, 256, 0, stream>>>(grids, Bg);

    // 16 row-tiles * 144 chunks = 2304 wave-tasks = 288 blocks of 8 waves
    kpowers<<<288, 256, 0, stream>>>(Aarr, Bg, trZ, partM, partS);

    // 256 batches, one wave each = 32 blocks of 8 waves
    kfinal<<<32, 256, 0, stream>>>(partM, partS, powerGT, out);
}